// KnnLoss_12841952215428
// MI455X (gfx1250) — compile-verified
//
#include <hip/hip_runtime.h>
#include <math.h>

typedef __attribute__((ext_vector_type(2))) float v2f;
typedef __attribute__((ext_vector_type(8))) float v8f;

#define NB    2
#define NPTS  8192
#define KSZ   16
#define KNN   8
#define R2F   0.01f
#define CT    32                     /* candidates per inner iteration (2 WMMA tiles) */
#define ITERS (NPTS / CT)            /* 256 */
#define WPB   8                      /* waves per block (256 threads) */

__global__ void __launch_bounds__(32) knn_zero_kernel(float* out) {
    if (threadIdx.x == 0) out[0] = 0.0f;
}

__global__ void __launch_bounds__(256) knn_loss_kernel(
        const float* __restrict__ pc, const float* __restrict__ mask,
        float* __restrict__ out) {
    __shared__ __align__(16) float d2lds[WPB][16][32];  // 16 KB: WMMA-layout -> row-layout transpose
    __shared__ __align__(16) float mdist[WPB][16][16];  // per-row 2-lane merge buffer
    __shared__ int                 midx [WPB][16][16];
    __shared__ float blockSum;

    const int tid  = threadIdx.x;
    const int wave = tid >> 5;
    const int lane = tid & 31;
    if (tid == 0) blockSum = 0.0f;
    __syncthreads();

    const int t     = blockIdx.x * WPB + wave;       // 0..1023 query tiles
    const int b     = t / (NPTS / 16);
    const int qbase = (t % (NPTS / 16)) * 16;
    const float* __restrict__ pcb = pc   + (size_t)b * NPTS * 3;
    const float* __restrict__ mb  = mask + (size_t)b * NPTS * KSZ;

    const int lo = lane & 15;
    const int hi = lane >> 4;

    // ---- A fragment (16x4 f32): lanes 0-15 -> (x,y)=K0,K1 ; lanes 16-31 -> (z,0)=K2,K3
    const float* qp = pcb + (size_t)(qbase + lo) * 3;
    const float qx = qp[0], qy = qp[1], qz = qp[2];
    v2f afrag;
    afrag.x = hi ? qz : qx;
    afrag.y = hi ? 0.0f : qy;

    // ---- per-lane squared norms of the 8 query rows this lane's C slice covers (M = v + 8*hi)
    float sq8[8];
#pragma unroll
    for (int v = 0; v < 8; ++v) {
        const float* p = pcb + (size_t)(qbase + v + 8 * hi) * 3;
        sq8[v] = p[0] * p[0] + p[1] * p[1] + p[2] * p[2];
    }

    // ---- top-8 state: lane owns row = lane/2, column half = (lane&1)*8
    const int row    = lane >> 1;
    const int half   = lane & 1;
    const int cbase0 = half * 8;
    const int self   = qbase + row;

    float bd[KNN];
    int   bi[KNN];
#pragma unroll
    for (int k = 0; k < KNN; ++k) { bd[k] = 3.0e38f; bi[k] = self; }

    // ---- software-pipelined candidate coordinate loads (L2-resident: pc is 96 KB/batch)
    const float* p0 = pcb + (size_t)(lo) * 3;
    const float* p1 = pcb + (size_t)(16 + lo) * 3;
    float c0x = p0[0], c0y = p0[1], c0z = p0[2];
    float c1x = p1[0], c1y = p1[1], c1z = p1[2];

    for (int j = 0; j < ITERS; ++j) {
        const int cb = j * CT;

        // prefetch next tile-pair (wraparound: branch-free; last prefetch is harmless)
        const int jn = (j + 1) & (ITERS - 1);
        const float* np0 = pcb + (size_t)(jn * CT + lo) * 3;
        const float* np1 = pcb + (size_t)(jn * CT + 16 + lo) * 3;
        const float n0x = np0[0], n0y = np0[1], n0z = np0[2];
        const float n1x = np1[0], n1y = np1[1], n1z = np1[2];

        // ---- B fragments (4x16 f32), same lane layout as A
        v2f b0, b1;
        b0.x = hi ? c0z : c0x;  b0.y = hi ? 0.0f : c0y;
        b1.x = hi ? c1z : c1x;  b1.y = hi ? 0.0f : c1y;

        v8f g0 = {};
        g0 = __builtin_amdgcn_wmma_f32_16x16x4_f32(false, afrag, false, b0,
                                                   (short)0, g0, false, false);
        v8f g1 = {};
        g1 = __builtin_amdgcn_wmma_f32_16x16x4_f32(false, afrag, false, b1,
                                                   (short)0, g1, false, false);

        const float sqc0 = c0x * c0x + c0y * c0y + c0z * c0z;
        const float sqc1 = c1x * c1x + c1y * c1y + c1z * c1z;

        // ---- d2 = ||q||^2 + ||c||^2 - 2*gram, clamp >= 0; scatter to row-major LDS
#pragma unroll
        for (int v = 0; v < 8; ++v) {
            const int m = v + 8 * hi;
            d2lds[wave][m][lo]      = fmaxf(fmaf(-2.0f, g0[v], sq8[v] + sqc0), 0.0f);
            d2lds[wave][m][16 + lo] = fmaxf(fmaf(-2.0f, g1[v], sq8[v] + sqc1), 0.0f);
        }
        __builtin_amdgcn_wave_barrier();   // intra-wave DS ops are in-order; pin the compiler

        // ---- lane reads its contiguous row-halves (4 x ds_load_b128)
        const float4 qa = *(const float4*)&d2lds[wave][row][cbase0];
        const float4 qb = *(const float4*)&d2lds[wave][row][cbase0 + 4];
        const float4 qc = *(const float4*)&d2lds[wave][row][16 + cbase0];
        const float4 qd = *(const float4*)&d2lds[wave][row][16 + cbase0 + 4];
        const float dv[16] = {qa.x, qa.y, qa.z, qa.w, qb.x, qb.y, qb.z, qb.w,
                              qc.x, qc.y, qc.z, qc.w, qd.x, qd.y, qd.z, qd.w};

#pragma unroll
        for (int u = 0; u < 16; ++u) {
            float nd = dv[u];
            int   ni = cb + cbase0 + ((u < 8) ? u : (u + 8));  // constant-folded per unroll step
            // rare-taken fast-path filter: single fused condition, then branchless insert
            if ((int)(nd <= R2F) & (int)(nd < bd[KNN - 1])) {
#pragma unroll
                for (int k = 0; k < KNN; ++k) {
                    const bool ins = nd < bd[k];
                    const float td = bd[k]; const int ti = bi[k];
                    bd[k] = ins ? nd : td;
                    bi[k] = ins ? ni : ti;
                    nd = ins ? td : nd;
                    ni = ins ? ti : ni;
                }
            }
        }
        __builtin_amdgcn_wave_barrier();

        c0x = n0x; c0y = n0y; c0z = n0z;
        c1x = n1x; c1y = n1y; c1z = n1z;
    }

    // ---- merge the two half-row lists per row through LDS
#pragma unroll
    for (int k = 0; k < KNN; ++k) {
        mdist[wave][row][cbase0 + k] = bd[k];
        midx [wave][row][cbase0 + k] = bi[k];
    }
    __builtin_amdgcn_wave_barrier();

    if (half == 0) {
        int fi[KNN];
        int pa = 0, pb = 0;
#pragma unroll
        for (int k = 0; k < KNN; ++k) {
            const int ja = pa < 7 ? pa : 7;
            const int jb = pb < 7 ? pb : 7;
            const float da = mdist[wave][row][ja];
            const float db = mdist[wave][row][8 + jb];
            const int   ia = midx [wave][row][ja];
            const int   ib = midx [wave][row][8 + jb];
            const int takeA = (int)(pb >= 8) |
                              ((int)(pa < 8) &
                               ((int)(da < db) | ((int)(da == db) & (int)(ia <= ib))));
            fi[k] = takeA ? ia : ib;
            pa += takeA;
            pb += 1 - takeA;
        }

        // ---- loss: sum_k sum_ks |mask[self][ks] - mask[fi[k]][ks]|  (self slots contribute 0)
        float m0[KSZ];
        const float* mr0 = mb + (size_t)self * KSZ;
#pragma unroll
        for (int s = 0; s < KSZ; ++s) m0[s] = mr0[s];

        float acc = 0.0f;
#pragma unroll
        for (int k = 0; k < KNN; ++k) {
            const int nk = fi[k];
            if (nk != self) {
                const float* mr = mb + (size_t)nk * KSZ;
#pragma unroll
                for (int s = 0; s < KSZ; ++s) acc += fabsf(m0[s] - mr[s]);
            }
        }
        atomicAdd(&blockSum, acc);
    }

    __syncthreads();
    if (tid == 0) {
        atomicAdd(out, blockSum * (1.0f / ((float)NB * (float)NPTS * (float)KNN)));
    }
}

extern "C" void kernel_launch(void* const* d_in, const int* in_sizes, int n_in,
                              void* d_out, int out_size, void* d_ws, size_t ws_size,
                              hipStream_t stream) {
    const float* pc   = (const float*)d_in[0];   // (2, 8192, 3) f32
    const float* mask = (const float*)d_in[1];   // (2, 8192, 16) f32
    float* out = (float*)d_out;                  // scalar f32
    (void)in_sizes; (void)n_in; (void)d_ws; (void)ws_size; (void)out_size;

    knn_zero_kernel<<<1, 32, 0, stream>>>(out);

    const int totalTiles = NB * (NPTS / 16);     // 1024 query tiles of 16
    const int blocks = totalTiles / WPB;         // 128 blocks x 256 threads
    knn_loss_kernel<<<blocks, 256, 0, stream>>>(pc, mask, out);
}